// ProposalTF_5970004541861
// MI455X (gfx1250) — compile-verified
//
#include <hip/hip_runtime.h>

// ---------------- problem constants ----------------
#define NB      8
#define NANCH   48000
#define NPAD    65536        // padded per-batch key count (power of 2)
#define PRE     2000         // PRE_NMS
#define PREPAD  2048
#define MWORDS  64           // 64 u32 words per mask row (2000 bits + pad)
#define CHUNK_ROWS 100       // rows per async LDS chunk in NMS kernel
#define NCHUNK  20           // 2000 / 100
#define IOU_THR 0.7f
#define IMG_H   800.0f
#define IMG_W   1280.0f

typedef unsigned int u32;
typedef unsigned long long u64;

// ---------------- kernel 1: score -> sort keys ----------------
__global__ void k_keys(const float* __restrict__ cls, u64* __restrict__ keys) {
    int t = blockIdx.x * blockDim.x + threadIdx.x;   // 0 .. NB*NPAD-1
    int b = t >> 16;
    int p = t & (NPAD - 1);
    u64 key = 0ull;
    if (p < NANCH) {
        float c0 = cls[(b * NANCH + p) * 2 + 0];
        float c1 = cls[(b * NANCH + p) * 2 + 1];
        float m  = fmaxf(c0, c1);
        float e0 = expf(c0 - m);
        float e1 = expf(c1 - m);
        float s  = e1 / (e0 + e1);                   // softmax[:,1]
        key = ((u64)__float_as_uint(s) << 32) | (u32)(~(u32)p);
    }
    keys[t] = key;
}

// ---------------- bitonic sort (descending per batch of 65536) ----------------
__global__ void k_sort_local(u64* __restrict__ keys) {
    __shared__ u64 s[2048];
    int tid = threadIdx.x;                           // 0..1023
    int base = blockIdx.x * 2048;                    // element base (chunk-aligned)
    s[tid]        = keys[base + tid];
    s[tid + 1024] = keys[base + tid + 1024];
    __syncthreads();
    for (int k = 2; k <= 2048; k <<= 1) {
        for (int j = k >> 1; j >= 1; j >>= 1) {
            int i  = 2 * tid - (tid & (j - 1));      // lower element of pair
            int gi = (base + i) & (NPAD - 1);        // per-batch index
            bool desc = (gi & k) == 0;
            u64 a = s[i], bb = s[i + j];
            if (desc ? (a < bb) : (a > bb)) { s[i] = bb; s[i + j] = a; }
            __syncthreads();
        }
    }
    keys[base + tid]        = s[tid];
    keys[base + tid + 1024] = s[tid + 1024];
}

__global__ void k_sort_global(u64* __restrict__ keys, int k, int j) {
    int t = blockIdx.x * blockDim.x + threadIdx.x;   // pair id, NB*NPAD/2 total
    int i = 2 * t - (t & (j - 1));
    int gi = i & (NPAD - 1);
    bool desc = (gi & k) == 0;
    u64 a = keys[i], b = keys[i + j];
    if (desc ? (a < b) : (a > b)) { keys[i] = b; keys[i + j] = a; }
}

__global__ void k_sort_merge_local(u64* __restrict__ keys, int k) {
    __shared__ u64 s[2048];
    int tid  = threadIdx.x;
    int base = blockIdx.x * 2048;
    s[tid]        = keys[base + tid];
    s[tid + 1024] = keys[base + tid + 1024];
    bool desc = (((base) & (NPAD - 1)) & k) == 0;    // constant per chunk (k>=4096)
    __syncthreads();
    for (int j = 1024; j >= 1; j >>= 1) {
        int i = 2 * tid - (tid & (j - 1));
        u64 a = s[i], bb = s[i + j];
        if (desc ? (a < bb) : (a > bb)) { s[i] = bb; s[i + j] = a; }
        __syncthreads();
    }
    keys[base + tid]        = s[tid];
    keys[base + tid + 1024] = s[tid + 1024];
}

// ---------------- kernel 3: gather top-2000, decode + clip ----------------
__global__ void k_gather(const u64* __restrict__ keys,
                         const float* __restrict__ enc,
                         float* __restrict__ topbox) {
    int t = blockIdx.x * blockDim.x + threadIdx.x;   // 0..2047
    int b = blockIdx.y;
    if (t >= PRE) return;
    u64 key = keys[b * NPAD + t];
    u32 p = ~((u32)key);                             // anchor index
    // anchor from flat index: p = (y*80 + x)*12 + a ; a = ratio_i*4 + scale_j
    u32 a = p % 12u, cell = p / 12u;
    u32 xg = cell % 80u, yg = cell / 80u;
    const float scales[4] = {0.25f, 0.5f, 1.0f, 2.0f};
    const float ratios[3] = {0.5f, 1.0f, 2.0f};
    float sc = scales[a & 3u];
    float rt = ratios[a >> 2];
    float sq = sqrtf(rt);
    float ha = sc / sq * 256.0f;
    float wa = sc * sq * 256.0f;
    float ya = (float)yg * 16.0f;
    float xa = (float)xg * 16.0f;
    const float* e = enc + ((size_t)b * NANCH + p) * 4;
    float ty = e[0] / 10.0f, tx = e[1] / 10.0f;
    float th = e[2] / 5.0f,  tw = e[3] / 5.0f;
    float h  = expf(th) * ha;
    float w  = expf(tw) * wa;
    float yc = ty * ha + ya;
    float xc = tx * wa + xa;
    float4 bx;
    bx.x = fminf(fmaxf(yc - h * 0.5f, 0.0f), IMG_H);
    bx.y = fminf(fmaxf(xc - w * 0.5f, 0.0f), IMG_W);
    bx.z = fminf(fmaxf(yc + h * 0.5f, 0.0f), IMG_H);
    bx.w = fminf(fmaxf(xc + w * 0.5f, 0.0f), IMG_W);
    ((float4*)topbox)[b * PREPAD + t] = bx;
}

// ---------------- kernel 4: suppression bitmask (parallel) ----------------
// block = 256 threads (8 waves); blockIdx.x = i-tile of 32 rows; blockIdx.y = batch
__global__ void k_mask(const float* __restrict__ topbox, u32* __restrict__ mask) {
    __shared__ float sy0[PRE], sx0[PRE], sy1[PRE], sx1[PRE];
    int b = blockIdx.y, tile = blockIdx.x;
    int tid = threadIdx.x;
    for (int t = tid; t < PRE; t += 256) {
        float4 bx = ((const float4*)topbox)[b * PREPAD + t];
        sy0[t] = bx.x; sx0[t] = bx.y; sy1[t] = bx.z; sx1[t] = bx.w;
    }
    __syncthreads();
    int lane = tid & 31;
    int wave = tid >> 5;
    int i = tile * 32 + lane;                        // row handled by this lane
    float iy0 = 0, ix0 = 0, iy1 = 0, ix1 = 0, ai = 0;
    if (i < PRE) {
        iy0 = sy0[i]; ix0 = sx0[i]; iy1 = sy1[i]; ix1 = sx1[i];
        ai = (iy1 - iy0) * (ix1 - ix0);
    }
    for (int w = wave * 8; w < wave * 8 + 8; ++w) {  // 8 words per wave
        u32 word = 0;
        int jbase = w * 32;
        #pragma unroll 4
        for (int bb = 0; bb < 32; ++bb) {
            int j = jbase + bb;
            if (j < PRE) {
                float jy0 = sy0[j], jx0 = sx0[j], jy1 = sy1[j], jx1 = sx1[j]; // LDS broadcast
                float aj = (jy1 - jy0) * (jx1 - jx0);
                float iy = fmaxf(fminf(iy1, jy1) - fmaxf(iy0, jy0), 0.0f);
                float ix = fmaxf(fminf(ix1, jx1) - fmaxf(ix0, jx0), 0.0f);
                float inter = iy * ix;
                float uni = ai + aj - inter;
                float iou = (uni > 0.0f) ? (inter / uni) : 0.0f;
                if (j > i && iou > IOU_THR) word |= (1u << bb);
            }
        }
        if (i < PRE) mask[(((size_t)b * PRE + i) << 6) + w] = word;
    }
}

// ---------------- kernel 5: sequential greedy NMS + output ----------------
// One wave32 per batch. Removed-set in 2 VGPRs/lane (63 words). Mask rows are
// streamed into LDS with double-buffered gfx1250 async global->LDS DMA so the
// 2000-step dependent chain only touches LDS + registers.
__global__ __launch_bounds__(32) void k_nms(const u32* __restrict__ mask,
                                            const float* __restrict__ topbox,
                                            float* __restrict__ out) {
    __shared__ u32 rowbuf[2][CHUNK_ROWS * MWORDS];   // 2 x 25.6 KB
    int b = blockIdx.x;
    int lane = threadIdx.x;
    const u32* mbase = mask + (size_t)b * PRE * MWORDS;
    u64 saddr = (u64)(size_t)mbase;
    u32 r0 = 0, r1 = 0;                              // removed words: lane, lane+32

    auto issue = [&](int c) {
        u32 gbytes  = (u32)c * (CHUNK_ROWS * MWORDS * 4);
        // low 32 bits of generic LDS pointer == wave-relative LDS byte address
        u32 ldsbase = (u32)(size_t)(void*)&rowbuf[c & 1][0];
        for (int it = 0; it < 50; ++it) {            // 50 * 32 lanes * 16B = 25600B
            u32 off   = (u32)(it * 32 + lane) * 16u;
            u32 voff  = gbytes + off;
            u32 laddr = ldsbase + off;
            asm volatile("global_load_async_to_lds_b128 %0, %1, %2"
                         :: "v"(laddr), "v"(voff), "s"(saddr) : "memory");
        }
    };

    issue(0);
    for (int c = 0; c < NCHUNK; ++c) {
        if (c + 1 < NCHUNK) {
            issue(c + 1);
            asm volatile("s_wait_asynccnt 50" ::: "memory"); // chunk c complete
        } else {
            asm volatile("s_wait_asynccnt 0" ::: "memory");
        }
        const u32* buf = &rowbuf[c & 1][0];
        #pragma unroll 4
        for (int il = 0; il < CHUNK_ROWS; ++il) {
            int i  = c * CHUNK_ROWS + il;
            int wi = i >> 5;                          // word holding bit i (uniform)
            u32 src = (wi < 32) ? r0 : r1;
            u32 wv  = (u32)__shfl((int)src, wi & 31, 32);
            if (!((wv >> (i & 31)) & 1u)) {           // keep[i] -> suppress row
                r0 |= buf[il * MWORDS + lane];
                r1 |= buf[il * MWORDS + 32 + lane];
            }
        }
    }

    // keep words (mask out padding bits)
    u32 k0 = ~r0;
    u32 k1 = ~r1;
    if (lane == 30) k1 &= 0xFFFFu;                    // word 62: j in [1984,2000)
    if (lane == 31) k1  = 0u;                         // word 63: invalid

    // zero all 100x4 outputs first (matches where(valid,...,0))
    float* ob = out + b * 100 * 4;
    for (int t = lane; t < 400; t += 32) ob[t] = 0.0f;
    asm volatile("s_wait_storecnt 0" ::: "memory");

    // in-order compaction of survivors; emit first 100
    int base = 0;
    for (int w = 0; w < 63 && base < 100; ++w) {
        u32 srck = (w < 32) ? k0 : k1;
        u32 bits = (u32)__shfl((int)srck, w & 31, 32);
        u32 mybit = (bits >> lane) & 1u;
        int pos = base + __popc(bits & ((1u << lane) - 1u));
        if (mybit && pos < 100) {
            int t = w * 32 + lane;
            float4 bx = ((const float4*)topbox)[b * PREPAD + t];
            ob[pos * 4 + 0] = bx.x / IMG_H;
            ob[pos * 4 + 1] = bx.y / IMG_W;
            ob[pos * 4 + 2] = bx.z / IMG_H;
            ob[pos * 4 + 3] = bx.w / IMG_W;
        }
        base += __popc(bits);
    }
}

// ---------------- host launcher ----------------
extern "C" void kernel_launch(void* const* d_in, const int* in_sizes, int n_in,
                              void* d_out, int out_size, void* d_ws, size_t ws_size,
                              hipStream_t stream) {
    (void)in_sizes; (void)n_in; (void)out_size; (void)ws_size;
    const float* enc = (const float*)d_in[1];        // box_encodings (B,N,4)
    const float* cls = (const float*)d_in[2];        // class_predictions (B,N,2)
    float* out = (float*)d_out;                      // (B,100,4)

    // ws layout: keys(4MB) | topbox(256KB) | mask(~4MB)  => ~8.6MB total
    u64*  keys   = (u64*)d_ws;
    float* topbox = (float*)(keys + (size_t)NB * NPAD);
    u32*  mask   = (u32*)(topbox + (size_t)NB * PREPAD * 4);

    // 1) scores -> sort keys (padded)
    k_keys<<<(NB * NPAD) / 256, 256, 0, stream>>>(cls, keys);

    // 2) bitonic sort, descending per batch
    k_sort_local<<<(NB * NPAD) / 2048, 1024, 0, stream>>>(keys);
    for (int k = 4096; k <= NPAD; k <<= 1) {
        for (int j = k >> 1; j >= 2048; j >>= 1)
            k_sort_global<<<(NB * NPAD / 2) / 256, 256, 0, stream>>>(keys, k, j);
        k_sort_merge_local<<<(NB * NPAD) / 2048, 1024, 0, stream>>>(keys, k);
    }

    // 3) gather + decode top-2000 boxes
    k_gather<<<dim3(PREPAD / 256, NB), 256, 0, stream>>>(keys, enc, topbox);

    // 4) parallel suppression bitmask
    k_mask<<<dim3((PRE + 31) / 32, NB), 256, 0, stream>>>(topbox, mask);

    // 5) serial scan (one wave per batch) + output
    k_nms<<<NB, 32, 0, stream>>>(mask, topbox, out);
}